// AutoMemoryModule_45638322487590
// MI455X (gfx1250) — compile-verified
//
#include <hip/hip_runtime.h>

typedef float v2f __attribute__((ext_vector_type(2)));
typedef float v8f __attribute__((ext_vector_type(8)));

#define MSL   256
#define EMB   1024
#define KTOT  (MSL * EMB)       // 262144
#define NBLK  256               // one block per token position
#define WPB   8                 // waves per block (256 threads, wave32)
#define KPW   (EMB / WPB)       // 128 K-elements per wave
#define NEGF  (-1e20f)

// ---------------------------------------------------------------------------
// Kernel 1: C[2x64] = A[2x262144] @ w1[262144x64].
// Block b owns token position b: its A-slice is exactly emb[tok_in(b)] and
// emb[tok_mem(b)] (K range 1024 == one token). Stage both rows in LDS once;
// inner loop is pure ds_load + 8-wide global clause + 4x WMMA, branch-free.
// ---------------------------------------------------------------------------
__global__ __launch_bounds__(256)
void k1_gemm_w1(const int* __restrict__ in_tok, int n_tok,
                const int* __restrict__ mem_tok,
                const float* __restrict__ emb,
                const float* __restrict__ w1,
                float* __restrict__ blk_part) {
  __shared__ float Alds[2][EMB];     // 8KB: the two embedding rows
  __shared__ float red[WPB][128];    // 4KB: per-wave partials

  const int tid = threadIdx.x;
  const int t   = blockIdx.x;                        // token position 0..255
  const int tokA = (t < n_tok) ? in_tok[t] : 0;      // pad with PAD=0
  const int tokB = mem_tok[t];
  {
    const float4* s0 = (const float4*)(emb + (size_t)tokA * EMB);
    const float4* s1 = (const float4*)(emb + (size_t)tokB * EMB);
    ((float4*)Alds[0])[tid] = s0[tid];               // 256 * 16B = 4KB each
    ((float4*)Alds[1])[tid] = s1[tid];
  }
  __syncthreads();

  const int  lane = tid & 31;
  const int  wid  = tid >> 5;
  const int  m    = lane & 15;        // M index (A) / N-in-tile (B)
  const int  half = lane >> 4;        // VGPR j, half h -> K = 2h + j
  const int  row  = (m < 2) ? m : 0;  // clamp so ds_load is always in range
  const bool act  = (m < 2);

  v8f c0 = {}; v8f c1 = {}; v8f c2 = {}; v8f c3 = {};

  const int kl0 = wid * KPW;
  #pragma unroll 4
  for (int kl = kl0; kl < kl0 + KPW; kl += 4) {
    const int kkl = kl + 2 * half;            // local K of this lane-half
    // A fragment from LDS; lanes m>=2 masked to zero (cndmask, no branch).
    const float a0 = Alds[row][kkl];
    const float a1 = Alds[row][kkl + 1];
    v2f a; a.x = act ? a0 : 0.f; a.y = act ? a1 : 0.f;
    // B fragments: w1 rows (t*1024 + kkl), (.. + 1); 4 N-tiles = 64 cols.
    const float* r0 = w1 + ((size_t)t * EMB + kkl) * 64;
    const float* r1 = r0 + 64;
    v2f b0; b0.x = r0[m +  0]; b0.y = r1[m +  0];
    v2f b1; b1.x = r0[m + 16]; b1.y = r1[m + 16];
    v2f b2; b2.x = r0[m + 32]; b2.y = r1[m + 32];
    v2f b3; b3.x = r0[m + 48]; b3.y = r1[m + 48];
    c0 = __builtin_amdgcn_wmma_f32_16x16x4_f32(false, a, false, b0, (short)0, c0, false, false);
    c1 = __builtin_amdgcn_wmma_f32_16x16x4_f32(false, a, false, b1, (short)0, c1, false, false);
    c2 = __builtin_amdgcn_wmma_f32_16x16x4_f32(false, a, false, b2, (short)0, c2, false, false);
    c3 = __builtin_amdgcn_wmma_f32_16x16x4_f32(false, a, false, b3, (short)0, c3, false, false);
  }

  // C layout: VGPR r, lanes 0-15 hold M=r. Rows M=0,1 are the live outputs.
  if (lane < 16) {
    red[wid][ 0 + lane] = c0[0];
    red[wid][16 + lane] = c1[0];
    red[wid][32 + lane] = c2[0];
    red[wid][48 + lane] = c3[0];
    red[wid][64 +  0 + lane] = c0[1];
    red[wid][64 + 16 + lane] = c1[1];
    red[wid][64 + 32 + lane] = c2[1];
    red[wid][64 + 48 + lane] = c3[1];
  }
  __syncthreads();
  if (tid < 128) {
    float s = 0.f;
    for (int w = 0; w < WPB; ++w) s += red[w][tid];
    blk_part[blockIdx.x * 128 + tid] = s;
  }
}

// ---------------------------------------------------------------------------
// Kernel 2: sum 256 block partials, + b1, ReLU; h[2x64] @ w2[64x256] + b2,
// sigmoid; apply PAD mask; emit 512 (token, score) candidates.
// ---------------------------------------------------------------------------
__global__ __launch_bounds__(256)
void k2_tail(const float* __restrict__ blk_part,
             const float* __restrict__ b1v,
             const float* __restrict__ w2,
             const float* __restrict__ b2v,
             const int* __restrict__ in_tok, int n_tok,
             const int* __restrict__ mem_tok,
             int* __restrict__ comb_tok,
             float* __restrict__ comb_sc) {
  __shared__ float h[128];   // [row(2)][64]
  const int tid = threadIdx.x;
  if (tid < 128) {
    float s = 0.f;
    for (int blk = 0; blk < NBLK; ++blk) s += blk_part[blk * 128 + tid];
    s += b1v[tid & 63];
    h[tid] = (s > 0.f) ? s : 0.f;   // ReLU
  }
  __syncthreads();
  float s0 = b2v[tid], s1 = b2v[tid];
  for (int e = 0; e < 64; ++e) {
    float w = w2[e * MSL + tid];
    s0 += h[e]      * w;
    s1 += h[64 + e] * w;
  }
  s0 = 1.f / (1.f + __expf(-s0));   // sigmoid
  s1 = 1.f / (1.f + __expf(-s1));
  const int ti = (tid < n_tok) ? in_tok[tid] : 0;
  const int tm = mem_tok[tid];
  comb_tok[tid]       = ti;
  comb_tok[MSL + tid] = tm;
  comb_sc[tid]        = (ti != 0) ? s0 : NEGF;
  comb_sc[MSL + tid]  = (tm != 0) ? s1 : NEGF;
}

// ---------------------------------------------------------------------------
// Kernel 3: only the 512 candidates beat NEG in the vocab table, so
// top-256-over-vocab == dedupe(max per token) + rank by (score desc, tok asc),
// remaining slots (PAD, NEG). O(512^2) in LDS, fully deterministic.
// ---------------------------------------------------------------------------
__global__ __launch_bounds__(512)
void k3_topk(const int* __restrict__ comb_tok,
             const float* __restrict__ comb_sc,
             float* __restrict__ out) {
  __shared__ int   toks[512];
  __shared__ float scs[512];
  __shared__ float maxs[512];
  __shared__ unsigned char keepA[512];
  const int tid = threadIdx.x;
  toks[tid] = comb_tok[tid];
  scs[tid]  = comb_sc[tid];
  if (tid < MSL) {
    out[tid]       = 0.0f;   // PAD token (as float)
    out[MSL + tid] = NEGF;   // score
  }
  __syncthreads();
  const int   mytok = toks[tid];
  float       mys   = scs[tid];
  bool        keep  = true;
  for (int j = 0; j < 512; ++j) {
    if (j == tid) continue;
    if (toks[j] == mytok) {
      if (scs[j] > mys) mys = scs[j];   // scatter-max per token
      if (j < tid) keep = false;        // dedupe: first occurrence wins
    }
  }
  if (!(mys > NEGF * 0.5f)) keep = false;  // `present` test
  maxs[tid]  = mys;
  keepA[tid] = keep ? 1 : 0;
  __syncthreads();
  if (keep) {
    int rank = 0;
    for (int j = 0; j < 512; ++j) {
      if (!keepA[j]) continue;
      if (maxs[j] > mys || (maxs[j] == mys && toks[j] < mytok)) ++rank;
    }
    if (rank < MSL) {
      out[rank]       = (float)mytok;
      out[MSL + rank] = mys;
    }
  }
}

// ---------------------------------------------------------------------------
extern "C" void kernel_launch(void* const* d_in, const int* in_sizes, int n_in,
                              void* d_out, int out_size, void* d_ws, size_t ws_size,
                              hipStream_t stream) {
  const int*   in_tok  = (const int*)  d_in[0];
  const int*   mem_tok = (const int*)  d_in[1];
  const float* emb     = (const float*)d_in[2];
  const float* w1      = (const float*)d_in[3];
  const float* b1v     = (const float*)d_in[4];
  const float* w2      = (const float*)d_in[5];
  const float* b2v     = (const float*)d_in[6];
  const int n_tok = in_sizes[0];   // 200 in the reference setup

  char*  ws       = (char*)d_ws;
  float* blk_part = (float*)ws;                              // 256*128 f32 = 128KB
  int*   comb_tok = (int*)  (ws + NBLK * 128 * sizeof(float));
  float* comb_sc  = (float*)(ws + NBLK * 128 * sizeof(float) + 512 * sizeof(int));

  k1_gemm_w1<<<NBLK, 256, 0, stream>>>(in_tok, n_tok, mem_tok, emb, w1, blk_part);
  k2_tail<<<1, 256, 0, stream>>>(blk_part, b1v, w2, b2v, in_tok, n_tok, mem_tok,
                                 comb_tok, comb_sc);
  k3_topk<<<1, 512, 0, stream>>>(comb_tok, comb_sc, (float*)d_out);
}